// Embedding_84645215470158
// MI455X (gfx1250) — compile-verified
//
#include <hip/hip_runtime.h>
#include <stdint.h>

// Embedding gather: out[r, :] = emb[ids[r], :], rows = B*S = 16384, EMBED_DIM = 1024 (fp32).
// Pure bandwidth problem (~128 MiB of traffic, ~5.5us at 23.3 TB/s). Uses the gfx1250
// async global<->LDS DMA path (ASYNCcnt): no VGPR data movement, 16 B128 ops in flight
// per wave. Output stores are non-temporal so the 128 MB table stays hot in the 192 MB L2.

#define WAVES_PER_BLOCK 4
#define BLOCK_THREADS   (WAVES_PER_BLOCK * 32)
#define EMBED_DIM       1024
#define ROW_BYTES       (EMBED_DIM * 4)          // 4096 B per row
#define CHUNK_BYTES     512                       // 32 lanes * 16 B per async B128 op
#define CHUNKS_PER_ROW  (ROW_BYTES / CHUNK_BYTES) // 8

#define CPOL_RT       0   // default temporal (loads: cache gathered rows / table in L2)
#define CPOL_STORE_NT 1   // TH[2:0]=1 -> non-temporal store (don't pollute L2 with output)

#if defined(__AMDGCN__)

#if __has_builtin(__builtin_amdgcn_global_load_async_to_lds_b128) && \
    __has_builtin(__builtin_amdgcn_global_store_async_from_lds_b128)
#define ASYNC_VIA_BUILTIN 1
#else
#define ASYNC_VIA_BUILTIN 0
#endif

typedef int v4i __attribute__((ext_vector_type(4)));
typedef v4i __attribute__((address_space(1))) as1_v4i;   // global
typedef v4i __attribute__((address_space(3))) as3_v4i;   // LDS
typedef void __attribute__((address_space(3))) as3_void;

__device__ __forceinline__ void async_load_16B(const void* g, void* l) {
#if ASYNC_VIA_BUILTIN
  __builtin_amdgcn_global_load_async_to_lds_b128((as1_v4i*)g, (as3_v4i*)l, 0, CPOL_RT);
#else
  uint32_t lds_addr = (uint32_t)(uintptr_t)(as3_void*)l;
  asm volatile("global_load_async_to_lds_b128 %0, %1, off"
               :: "v"(lds_addr), "v"((uint64_t)(uintptr_t)g) : "memory");
#endif
}

__device__ __forceinline__ void async_store_16B(void* g, const void* l) {
#if ASYNC_VIA_BUILTIN
  __builtin_amdgcn_global_store_async_from_lds_b128((as1_v4i*)g, (as3_v4i*)l, 0,
                                                    CPOL_STORE_NT);
#else
  uint32_t lds_addr = (uint32_t)(uintptr_t)(as3_void*)l;
  asm volatile("global_store_async_from_lds_b128 %0, %1, off th:TH_STORE_NT"
               :: "v"((uint64_t)(uintptr_t)g), "v"(lds_addr) : "memory");
#endif
}

__device__ __forceinline__ void wait_async_0() {
#if __has_builtin(__builtin_amdgcn_s_wait_asynccnt)
  __builtin_amdgcn_s_wait_asynccnt(0);
#else
  asm volatile("s_wait_asynccnt 0x0" ::: "memory");
#endif
}

#endif // __AMDGCN__

__global__ __launch_bounds__(BLOCK_THREADS) void embed_gather_async(
    const int* __restrict__ ids, const float* __restrict__ emb,
    float* __restrict__ out, int nrows)
{
#if defined(__AMDGCN__)
  // 4 waves * 2 buffers * 4 KiB = 32 KiB LDS per block -> ~10 blocks/WGP by LDS.
  __shared__ char smem[WAVES_PER_BLOCK * 2 * ROW_BYTES];

  const int lane  = threadIdx.x & 31;
  const int wave  = threadIdx.x >> 5;
  const int gwave = blockIdx.x * WAVES_PER_BLOCK + wave;
  const int nwav  = gridDim.x * WAVES_PER_BLOCK;

  char* wbuf = smem + wave * (2 * ROW_BYTES);
  const int base = lane * 16;

  int it = 0;
  for (int r = gwave; r < nrows; r += nwav, ++it) {
    char* buf = wbuf + (it & 1) * ROW_BYTES;          // double buffer
    const int tok = __builtin_amdgcn_readfirstlane(ids[r]);
    const char* src = (const char*)(emb + (size_t)tok * EMBED_DIM);
    char* dst = (char*)(out + (size_t)r * EMBED_DIM);

    // Issue 8 async loads (4 KiB row -> LDS). These overlap with the previous
    // row's still-outstanding async stores (which read the *other* buffer).
#pragma unroll
    for (int c = 0; c < CHUNKS_PER_ROW; ++c)
      async_load_16B(src + base + c * CHUNK_BYTES, buf + base + c * CHUNK_BYTES);

    // ASYNCcnt is a single counter and load/store completions are mutually
    // unordered, so only a wait-to-zero safely orders LDS write -> LDS read.
    // It also guarantees stores of row (it-1) finished before row (it+1)
    // reuses that buffer (enforced one iteration early).
    wait_async_0();

    // Issue 8 async NT stores (LDS -> 4 KiB output row); left in flight into
    // the next iteration's load phase.
#pragma unroll
    for (int c = 0; c < CHUNKS_PER_ROW; ++c)
      async_store_16B(dst + base + c * CHUNK_BYTES, buf + base + c * CHUNK_BYTES);
  }
  wait_async_0();   // drain before wave exit
#endif
}

extern "C" void kernel_launch(void* const* d_in, const int* in_sizes, int n_in,
                              void* d_out, int out_size, void* d_ws, size_t ws_size,
                              hipStream_t stream) {
  const int*   ids = (const int*)d_in[0];    // token_ids, int32, B*S = 16384
  const float* emb = (const float*)d_in[1];  // embedding, fp32, 32000 x 1024
  float*       out = (float*)d_out;          // fp32, 16384 x 1024
  (void)d_ws; (void)ws_size; (void)n_in; (void)out_size;

  int nrows = in_sizes[0];
  int blocks = (nrows + WAVES_PER_BLOCK - 1) / WAVES_PER_BLOCK;
  if (blocks > 1024) blocks = 1024;          // ~4 rows/wave -> pipeline cycles
  embed_gather_async<<<blocks, BLOCK_THREADS, 0, stream>>>(ids, emb, out, nrows);
}